// MHC_43757126811708
// MI455X (gfx1250) — compile-verified
//
#include <hip/hip_runtime.h>
#include <hip/hip_bf16.h>
#include <math.h>

// ---------------------------------------------------------------------------
// mhc (multi-head coupling) fused kernel for MI455X / gfx1250
//   x = RMSNorm(permute(stream))  ;  H = x @ Wcat.T  (24 cols, padded to 32)
//   Since RMSNorm is a per-row scalar, fuse: dot(x_raw, W) * rsqrt(mean(x^2)).
//   GEMM M=8192, N=24->32, K=8192 via V_WMMA_F32_16X16X4_F32 (fp32 path;
//   problem is HBM-bound: 256MB @ 23.3TB/s ~= 11us, WMMA f32 keeps compute
//   under that roof).
// ---------------------------------------------------------------------------

typedef float v2f __attribute__((ext_vector_type(2)));
typedef float v8f __attribute__((ext_vector_type(8)));

#define T_LEN   8192
#define D_HID   2048
#define ND      8192
#define EPS_RMS 1e-8f
#define SINK_IT 20

// ---------------------------------------------------------------------------
// Pack W_pre(4,8192) / W_post(4,8192) / W_res(16,8192) into WMMA B-fragment
// order, N padded to 32 (2 tiles of 16). Element for (k-block kb, tile nt,
// lane L): column c = nt*16 + (L&15), k = kb*4 + (L>>4)*2 + {0,1}.
// wpack index = (kb*2 + nt)*32 + L   (float2 elements, 1 MB total)
// ---------------------------------------------------------------------------
__global__ __launch_bounds__(256) void mhc_pack_w(
    const float* __restrict__ Wpre,
    const float* __restrict__ Wpost,
    const float* __restrict__ Wres,
    float2* __restrict__ wpack) {
  int gid = blockIdx.x * 256 + threadIdx.x;      // 0 .. 2048*64-1
  if (gid >= 2048 * 64) return;
  int kb  = gid >> 6;
  int rem = gid & 63;
  int nt  = rem >> 5;
  int L   = rem & 31;
  int c   = nt * 16 + (L & 15);
  int kk  = kb * 4 + ((L >> 4) << 1);
  float2 v; v.x = 0.0f; v.y = 0.0f;
  if (c < 24) {
    const float* src;
    if (c < 4)       src = Wpre  + (size_t)c * ND;
    else if (c < 8)  src = Wpost + (size_t)(c - 4) * ND;
    else             src = Wres  + (size_t)(c - 8) * ND;
    v.x = src[kk];
    v.y = src[kk + 1];
  }
  wpack[gid] = v;
}

// ---------------------------------------------------------------------------
// Main fused kernel: one block = 16-row M-tile, 8 waves split K (1024 each).
// ---------------------------------------------------------------------------
__global__ __launch_bounds__(256) void mhc_main(
    const float* __restrict__ stream_in,   // (4, 8192, 2048)
    const float2* __restrict__ wpack,      // packed B fragments
    const float* __restrict__ b_pre,       // (4,)
    const float* __restrict__ b_post,      // (4,)
    const float* __restrict__ b_res,       // (4,4) flat
    const float* __restrict__ p_alpha_pre,
    const float* __restrict__ p_alpha_post,
    const float* __restrict__ p_alpha_res,
    float* __restrict__ out) {
  __shared__ float lds_acc[512];   // 2 tiles x 16x16 partial C
  __shared__ float lds_ss[16];     // per-row sum of squares

  const int tid   = threadIdx.x;
  const int wave  = tid >> 5;
  const int lane  = tid & 31;
  const int t0    = blockIdx.x << 4;     // 16 rows per block
  const int m     = lane & 15;           // M row within tile (A), N col (B/C)
  const int khalf = lane >> 4;           // K pair select for A/B fragments

  for (int i = tid; i < 512; i += 256) lds_acc[i] = 0.0f;
  if (tid < 16) lds_ss[tid] = 0.0f;
  __syncthreads();

  // --- this wave's contiguous K chunk [kbeg, kbeg+1024) -------------------
  const int kbeg = wave << 10;           // 0,1024,...,7168
  const int seg  = kbeg >> 11;           // stream segment (0..3)
  const int j0   = kbeg & 2047;          // offset inside segment
  const float* aptr = stream_in + (size_t)seg * T_LEN * D_HID
                    + (size_t)(t0 + m) * D_HID + j0 + khalf * 2;
  const size_t kb0 = (size_t)(kbeg >> 2);          // first k-block
  const float2* bptr = wpack + kb0 * 64 + lane;    // (kb*2+nt)*32 + lane

  v8f c0 = {};
  v8f c1 = {};
  float ss = 0.0f;

#pragma unroll 4
  for (int kb = 0; kb < 256; ++kb) {
    const float2 av  = *(const float2*)(aptr + (size_t)kb * 4);
    const float2 b0v = bptr[(size_t)kb * 64];        // N-tile 0
    const float2 b1v = bptr[(size_t)kb * 64 + 32];   // N-tile 1
    v2f a;  a.x  = av.x;  a.y  = av.y;
    v2f b0; b0.x = b0v.x; b0.y = b0v.y;
    v2f b1; b1.x = b1v.x; b1.y = b1v.y;
    // D = A(16x4) * B(4x16) + C  -- fp32 WMMA, wave32
    c0 = __builtin_amdgcn_wmma_f32_16x16x4_f32(false, a, false, b0,
                                               (short)0, c0, false, false);
    c1 = __builtin_amdgcn_wmma_f32_16x16x4_f32(false, a, false, b1,
                                               (short)0, c1, false, false);
    ss = fmaf(av.x, av.x, ss);
    ss = fmaf(av.y, av.y, ss);
  }

  // --- cross-wave reduction in LDS ----------------------------------------
  // C/D layout: VGPR r holds row M = r + (lane<16 ? 0 : 8), col N = lane&15.
  const int rowbase = (lane < 16) ? 0 : 8;
#pragma unroll
  for (int r = 0; r < 8; ++r) {
    const int Mrow = rowbase + r;
    atomicAdd(&lds_acc[Mrow * 16 + m],       c0[r]);
    atomicAdd(&lds_acc[256 + Mrow * 16 + m], c1[r]);
  }
  atomicAdd(&lds_ss[m], ss);   // lane covers row m's K-subset
  __syncthreads();

  // --- epilogue: 16 lanes, one output row each ----------------------------
  if (tid < 16) {
    const int t = t0 + tid;
    const float a_pre  = p_alpha_pre[0];
    const float a_post = p_alpha_post[0];
    const float a_res  = p_alpha_res[0];

    const float rinv = rsqrtf(lds_ss[tid] * (1.0f / 8192.0f) + EPS_RMS);

    float dots[24];
#pragma unroll
    for (int c = 0; c < 16; ++c) dots[c] = lds_acc[tid * 16 + c];
#pragma unroll
    for (int c = 0; c < 8; ++c)  dots[16 + c] = lds_acc[256 + tid * 16 + c];

    float* out_res  = out;                             // (T,4,4)
    float* out_pre  = out + (size_t)T_LEN * 16;        // (T,4)
    float* out_post = out + (size_t)T_LEN * 20;        // (T,4)

#pragma unroll
    for (int c = 0; c < 4; ++c) {
      const float zp = a_pre * dots[c] * rinv + b_pre[c];
      out_pre[t * 4 + c] = 1.0f / (1.0f + __expf(-zp));
      const float zq = a_post * dots[4 + c] * rinv + b_post[c];
      out_post[t * 4 + c] = 2.0f / (1.0f + __expf(-zq));
    }

    // Sinkhorn-Knopp on the 4x4 block
    float Mm[16];
#pragma unroll
    for (int e = 0; e < 16; ++e)
      Mm[e] = __expf(a_res * dots[8 + e] * rinv + b_res[e]);
#pragma unroll 1
    for (int it = 0; it < SINK_IT; ++it) {
#pragma unroll
      for (int rr = 0; rr < 4; ++rr) {
        const float inv = 1.0f / (Mm[rr*4] + Mm[rr*4+1] + Mm[rr*4+2] + Mm[rr*4+3]);
        Mm[rr*4]   *= inv; Mm[rr*4+1] *= inv;
        Mm[rr*4+2] *= inv; Mm[rr*4+3] *= inv;
      }
#pragma unroll
      for (int cc = 0; cc < 4; ++cc) {
        const float inv = 1.0f / (Mm[cc] + Mm[4+cc] + Mm[8+cc] + Mm[12+cc]);
        Mm[cc] *= inv; Mm[4+cc] *= inv; Mm[8+cc] *= inv; Mm[12+cc] *= inv;
      }
    }
#pragma unroll
    for (int e = 0; e < 16; ++e) out_res[t * 16 + e] = Mm[e];
  }
}

// ---------------------------------------------------------------------------
extern "C" void kernel_launch(void* const* d_in, const int* in_sizes, int n_in,
                              void* d_out, int out_size, void* d_ws, size_t ws_size,
                              hipStream_t stream) {
  const float* stream_in = (const float*)d_in[0];
  const float* W_pre     = (const float*)d_in[1];
  const float* W_post    = (const float*)d_in[2];
  const float* W_res     = (const float*)d_in[3];
  const float* b_pre     = (const float*)d_in[4];
  const float* b_post    = (const float*)d_in[5];
  const float* b_res     = (const float*)d_in[6];
  const float* a_pre     = (const float*)d_in[7];
  const float* a_post    = (const float*)d_in[8];
  const float* a_res     = (const float*)d_in[9];

  float2* wpack = (float2*)d_ws;   // needs 2048*64*8 B = 1 MB of scratch

  // 2048 k-blocks * 2 N-tiles * 32 lanes = 131072 pack elements
  mhc_pack_w<<<512, 256, 0, stream>>>(W_pre, W_post, W_res, wpack);

  // 8192 rows / 16 per block = 512 blocks, 8 waves each (K split)
  mhc_main<<<512, 256, 0, stream>>>(stream_in, wpack, b_pre, b_post, b_res,
                                    a_pre, a_post, a_res, (float*)d_out);
}